// MoV3D_38972533243928
// MI455X (gfx1250) — compile-verified
//
#include <hip/hip_runtime.h>

#define HH 48
#define WW 48
#define DDIM 48
#define CC 32
#define NH 4
#define HD 8
#define AMPC 2
#define KKN 5
#define KZ 56    // zi = zg + 4, zg in [-2,49] -> zi in [2,54); pad row to 56
#define TZ 12    // voxels per wave tile (band m..m+4 fits in one 16x16 tile)
#define SSTR 21  // stage row stride (8*21 mod 64 = 40 -> conflict-free halves)
#define NTH 128

typedef float v2f __attribute__((ext_vector_type(2)));
typedef float v4f __attribute__((ext_vector_type(4)));
typedef float v8f __attribute__((ext_vector_type(8)));

__device__ __forceinline__ v8f wmma4(v2f a, v2f b, v8f c) {
  // V_WMMA_F32_16X16X4_F32 : D(16x16,f32) = A(16x4,f32) * B(4x16,f32) + C
  return __builtin_amdgcn_wmma_f32_16x16x4_f32(false, a, false, b, (short)0, c,
                                               false, false);
}

__global__ __launch_bounds__(NTH) void na3d_wmma_kernel(
    const float* __restrict__ xq, const float* __restrict__ xk,
    float* __restrict__ out) {
  __shared__ float q_lds[HD][DDIM];      // [c][z], scaled by 8^-0.5 * log2(e)
  __shared__ float k_lds[25 * HD * KZ];  // [xy-line][c][zi]
  __shared__ float s_stage[4][16 * SSTR];

  const int bid = blockIdx.x;
  const int b = bid / (HH * WW);
  const int xy = bid % (HH * WW);
  const int x = xy / WW;
  const int y = xy % WW;

  const int tid = threadIdx.x;
  const int wave = tid >> 5;
  const int lane = tid & 31;
  const int half = lane >> 4;  // 0/1 selects K pair of WMMA fragments
  const int m = lane & 15;     // tile row (voxel) / tile column index
  const int z0 = wave * TZ;
  const int zg = z0 + (m < TZ ? m : TZ - 1);  // clamped voxel z (rows >=12 unused)

  // 8^-0.5 * log2(e): scores produced directly in log2 domain -> exp2 softmax
  const float scale = 0.35355339059327379f * 1.4426950408889634f;
  const float NEGINF = -__builtin_inff();

  // one-time: zero always-OOB z-edge columns (zg in {-2,-1,48,49})
  for (int idx = tid; idx < 25 * HD * 2; idx += NTH) {
    int row = idx >> 1;
    int zi = (idx & 1) ? 52 : 2;
    v2f zz = {0.f, 0.f};
    *(v2f*)&k_lds[row * KZ + zi] = zz;
  }
  // one-time: zero the stage (padding cols stay 0 forever -> no NaN reads)
  for (int idx = tid; idx < 4 * 16 * SSTR; idx += NTH)
    ((float*)s_stage)[idx] = 0.f;

  // per-lane tap set: half0 -> t in {0,1,2}, half1 -> t in {3,4,(5=dummy)}
  float bz[3], oz[3];
#pragma unroll
  for (int tt = 0; tt < 3; ++tt) {
    const int t = 3 * half + tt;
    const bool okz = (t < 5) && ((unsigned)(zg + t - AMPC) < DDIM);
    bz[tt] = okz ? 0.f : NEGINF;  // additive mask (branchless)
    oz[tt] = (float)(t - AMPC);
  }

  float acc0 = 0.f, acc1 = 0.f, acc2 = 0.f;  // sum over heads of v_d

  for (int h = 0; h < NH; ++h) {
    __syncthreads();
    // ---- q line: 96 x float4 (rows are 16B aligned) ----
    if (tid < 96) {
      const int c = tid / 12, q4 = tid % 12;
      v4f v = *(const v4f*)&xq[(((size_t)b * CC + (c * NH + h)) * HH + x) *
                                   (WW * DDIM) +
                               (size_t)y * DDIM + 4 * q4];
      v4f s = {v.x * scale, v.y * scale, v.z * scale, v.w * scale};
      *(v4f*)&q_lds[c][4 * q4] = s;
    }
    // ---- k 5x5 xy-window, in-range z part: 2400 x float4 ----
    for (int idx = tid; idx < 25 * HD * 12; idx += NTH) {
      const int line = idx / (HD * 12);
      const int rem = idx % (HD * 12);
      const int c = rem / 12;
      const int q4 = rem % 12;
      const int xg = x + line / KKN - AMPC;
      const int yg = y + line % KKN - AMPC;
      v4f v = {0.f, 0.f, 0.f, 0.f};
      if ((unsigned)xg < HH && (unsigned)yg < WW)
        v = *(const v4f*)&xk[(((size_t)b * CC + (c * NH + h)) * HH + xg) *
                                 (WW * DDIM) +
                             (size_t)yg * DDIM + 4 * q4];
      *(v4f*)&k_lds[(line * HD + c) * KZ + 4 + 4 * q4] = v;
    }
    __syncthreads();

    // A fragment: A[M=voxel][K=c]; lane m = M, (half,vgpr) = K
    v2f Alo, Ahi;
    Alo.x = q_lds[2 * half + 0][zg];
    Alo.y = q_lds[2 * half + 1][zg];
    Ahi.x = q_lds[4 + 2 * half + 0][zg];
    Ahi.y = q_lds[4 + 2 * half + 1][zg];

    float sm = -1.0e30f, ss = 0.f, sa0 = 0.f, sa1 = 0.f, sa2 = 0.f;
    float* srow = s_stage[wave];
    const int zb = z0 + 2 + m;  // zi of tile column m (col j -> zg = z0+j-2)

    for (int di = 0; di < KKN; ++di) {
      const bool vx = (unsigned)(x + di - AMPC) < HH;
      const float ox = (float)(di - AMPC);
      for (int dj = 0; dj < KKN; ++dj) {
        const bool vxy = vx && ((unsigned)(y + dj - AMPC) < WW);
        const float bxy = vxy ? 0.f : NEGINF;  // uniform additive mask
        const float oy = (float)(dj - AMPC);
        const float* kc = &k_lds[(di * KKN + dj) * HD * KZ];

        // B fragment: B[K=c][N=col]; lane m = N, (half,vgpr) = K
        v2f Blo, Bhi;
        Blo.x = kc[(2 * half + 0) * KZ + zb];
        Blo.y = kc[(2 * half + 1) * KZ + zb];
        Bhi.x = kc[(4 + 2 * half + 0) * KZ + zb];
        Bhi.y = kc[(4 + 2 * half + 1) * KZ + zb];

        v8f D = {};
        D = wmma4(Alo, Blo, D);  // channels 0..3
        D = wmma4(Ahi, Bhi, D);  // channels 4..7

        // stage full 16x16 tile (branchless)
#pragma unroll
        for (int v = 0; v < 8; ++v) srow[(v + 8 * half) * SSTR + m] = D[v];

        // band gather + split online softmax; additive -inf masks keep the
        // loads unconditional (no scalar branch diamonds)
#pragma unroll
        for (int tt = 0; tt < 3; ++tt) {
          const int t = 3 * half + tt;
          const float val = srow[m * SSTR + m + t] + bxy + bz[tt];
          const float nm = fmaxf(sm, val);
          const float es = exp2f(sm - nm);
          const float ev = exp2f(val - nm);
          ss = ss * es + ev;
          sa0 = sa0 * es + ev * ox;
          sa1 = sa1 * es + ev * oy;
          sa2 = sa2 * es + ev * oz[tt];
          sm = nm;
        }
      }
    }

    // merge the two half-wave partial softmax states (lane <-> lane^16)
    const float om = __shfl_xor(sm, 16, 32);
    const float os = __shfl_xor(ss, 16, 32);
    const float ob0 = __shfl_xor(sa0, 16, 32);
    const float ob1 = __shfl_xor(sa1, 16, 32);
    const float ob2 = __shfl_xor(sa2, 16, 32);
    const float nm = fmaxf(sm, om);
    const float e1 = exp2f(sm - nm);
    const float e2 = exp2f(om - nm);
    const float inv = 1.0f / (ss * e1 + os * e2);
    acc0 += (sa0 * e1 + ob0 * e2) * inv;
    acc1 += (sa1 * e1 + ob1 * e2) * inv;
    acc2 += (sa2 * e1 + ob2 * e2) * inv;
  }

  if (half == 0 && m < TZ) {
    const float r = 1.0f / (float)NH;
    const size_t plane = (size_t)HH * WW * DDIM;
    const size_t vox = (size_t)x * (WW * DDIM) + (size_t)y * DDIM + (z0 + m);
    const size_t base = (size_t)b * 3 * plane;
    out[base + 0 * plane + vox] = acc0 * r;
    out[base + 1 * plane + vox] = acc1 * r;
    out[base + 2 * plane + vox] = acc2 * r;
  }
}

extern "C" void kernel_launch(void* const* d_in, const int* in_sizes, int n_in,
                              void* d_out, int out_size, void* d_ws,
                              size_t ws_size, hipStream_t stream) {
  const float* xq = (const float*)d_in[0];
  const float* xk = (const float*)d_in[1];
  float* out = (float*)d_out;
  const int B = in_sizes[0] / (CC * HH * WW * DDIM);  // = 2
  dim3 grid((unsigned)(B * HH * WW));
  na3d_wmma_kernel<<<grid, NTH, 0, stream>>>(xq, xk, out);
}